// VectorQuantizer_17136919511056
// MI455X (gfx1250) — compile-verified
//
#include <hip/hip_runtime.h>
#include <hip/hip_bf16.h>

// Problem constants (match reference)
#define N_PTS   262144
#define K_EMB   1024
#define D_EMB   64
#define LDS_STRIDE 68                 // 64 + 4 pad -> conflict-free ds_load_b64
#define W2OFF   (K_EMB * LDS_STRIDE)  // w^2 table after padded codebook

typedef __attribute__((ext_vector_type(2))) float v2f;
typedef __attribute__((ext_vector_type(8))) float v8f;

// ---------------------------------------------------------------------------
// Kernel A: fp32 WMMA distance GEMM + per-row argmin over 1024 codes.
// One wave owns a 16-row block; LDS holds the whole (padded) codebook + w2.
// C-tile layout (v_wmma_f32_16x16x4_f32): VGPR j, lanes 0-15 -> row m0+j,
// lanes 16-31 -> row m0+8+j; column = tile_base + (lane & 15).
// ---------------------------------------------------------------------------
__global__ __launch_bounds__(512, 1)
void vq_argmin_kernel(const float* __restrict__ x,
                      const float* __restrict__ w,
                      int* __restrict__ idxbuf) {
    extern __shared__ float lds[];
    const int tid = threadIdx.x;

    // Stage codebook into padded LDS
    for (int t = tid; t < K_EMB * D_EMB; t += 512) {
        int r = t >> 6, c = t & 63;
        lds[r * LDS_STRIDE + c] = w[t];
    }
    __syncthreads();
    // Per-code squared norms
    for (int k = tid; k < K_EMB; k += 512) {
        float s = 0.f;
        #pragma unroll
        for (int d = 0; d < D_EMB; ++d) {
            float v = lds[k * LDS_STRIDE + d];
            s += v * v;
        }
        lds[W2OFF + k] = s;
    }
    __syncthreads();

    const int lane  = tid & 31;
    const int col16 = lane & 15;          // column slot within 16-wide tile
    const int kd    = (lane >> 4) << 1;   // K-dim sub-offset (0 or 2)
    const int gwave = blockIdx.x * 16 + (tid >> 5);
    const int totalWaves = gridDim.x * 16;

    for (int rb = gwave; rb < (N_PTS / 16); rb += totalWaves) {
        const int m0 = rb << 4;
        // Load A fragments once: lane holds x[m0+col16][4*s+kd .. +1]
        v2f areg[16];
        const float* arow = x + (size_t)(m0 + col16) * D_EMB;
        #pragma unroll
        for (int s = 0; s < 16; ++s)
            areg[s] = *(const v2f*)(arow + 4 * s + kd);

        float minv[8];
        int   mini[8];
        #pragma unroll
        for (int j = 0; j < 8; ++j) { minv[j] = 3.4e38f; mini[j] = 0; }

        for (int k0 = 0; k0 < K_EMB; k0 += 32) {
            const int kA = k0 + col16;
            const int kB = k0 + 16 + col16;
            v8f cA = {0.f,0.f,0.f,0.f,0.f,0.f,0.f,0.f};
            v8f cB = {0.f,0.f,0.f,0.f,0.f,0.f,0.f,0.f};
            #pragma unroll
            for (int s = 0; s < 16; ++s) {
                v2f bA = *(const v2f*)(&lds[kA * LDS_STRIDE + 4 * s + kd]);
                v2f bB = *(const v2f*)(&lds[kB * LDS_STRIDE + 4 * s + kd]);
                cA = __builtin_amdgcn_wmma_f32_16x16x4_f32(
                        false, areg[s], false, bA, (short)0, cA, false, false);
                cB = __builtin_amdgcn_wmma_f32_16x16x4_f32(
                        false, areg[s], false, bB, (short)0, cB, false, false);
            }
            const float w2A = lds[W2OFF + kA];
            const float w2B = lds[W2OFF + kB];
            #pragma unroll
            for (int j = 0; j < 8; ++j) {
                float sA = w2A - 2.0f * cA[j];
                if (sA < minv[j]) { minv[j] = sA; mini[j] = kA; }
                float sB = w2B - 2.0f * cB[j];
                if (sB < minv[j]) { minv[j] = sB; mini[j] = kB; }
            }
        }

        // Cross-lane argmin within each 16-lane half (xor masks <16 stay in-half)
        #pragma unroll
        for (int j = 0; j < 8; ++j) {
            float v = minv[j];
            int   ix = mini[j];
            #pragma unroll
            for (int m = 8; m >= 1; m >>= 1) {
                float ov  = __shfl_xor(v, m, 32);
                int   oix = __shfl_xor(ix, m, 32);
                if (ov < v || (ov == v && oix < ix)) { v = ov; ix = oix; }
            }
            minv[j] = v; mini[j] = ix;
        }
        if (col16 == 0) {
            const int base = m0 + ((lane >> 4) << 3);
            #pragma unroll
            for (int j = 0; j < 8; ++j) idxbuf[base + j] = mini[j];
        }
    }
}

// ---------------------------------------------------------------------------
// Kernel B: quantized_st = emb[idx]  + deterministic partial loss sums.
// Scalar b32 stores (output base is offset by 1 float -> not 16B aligned).
// ---------------------------------------------------------------------------
__global__ __launch_bounds__(256)
void vq_quant_loss_kernel(const float* __restrict__ x,
                          const float* __restrict__ w,
                          const int* __restrict__ idxbuf,
                          float* __restrict__ outq,
                          float* __restrict__ partials) {
    __shared__ float red[256];
    const size_t total  = (size_t)N_PTS * D_EMB;
    const size_t stride = (size_t)gridDim.x * blockDim.x;
    float acc = 0.f;
    for (size_t i = (size_t)blockIdx.x * blockDim.x + threadIdx.x;
         i < total; i += stride) {
        const int n = (int)(i >> 6);
        const int d = (int)(i & 63);
        const float q  = w[(size_t)idxbuf[n] * D_EMB + d];
        const float xv = x[i];
        outq[i] = q;
        const float df = q - xv;
        acc += df * df;
    }
    red[threadIdx.x] = acc;
    __syncthreads();
    for (int s = 128; s > 0; s >>= 1) {
        if (threadIdx.x < s) red[threadIdx.x] += red[threadIdx.x + s];
        __syncthreads();
    }
    if (threadIdx.x == 0) partials[blockIdx.x] = red[0];
}

// Kernel C: deterministic serial sum of fixed-count partials -> loss scalar
__global__ void vq_finalize_kernel(const float* __restrict__ partials,
                                   int nPart, float* __restrict__ out0) {
    if (blockIdx.x == 0 && threadIdx.x == 0) {
        float s = 0.f;
        for (int i = 0; i < nPart; ++i) s += partials[i];
        out0[0] = 1.25f * s / (float)((size_t)N_PTS * D_EMB);
    }
}

// Kernel D: one-hot encodings fill (1.07 GB write -> the HBM-bound phase)
__global__ __launch_bounds__(256)
void vq_encodings_kernel(const int* __restrict__ idxbuf,
                         float* __restrict__ enc) {
    const size_t total  = (size_t)N_PTS * K_EMB;
    const size_t stride = (size_t)gridDim.x * blockDim.x;
    for (size_t i = (size_t)blockIdx.x * blockDim.x + threadIdx.x;
         i < total; i += stride) {
        const int n = (int)(i >> 10);
        const int k = (int)(i & 1023);
        enc[i] = (k == idxbuf[n]) ? 1.0f : 0.0f;
    }
}

extern "C" void kernel_launch(void* const* d_in, const int* in_sizes, int n_in,
                              void* d_out, int out_size, void* d_ws, size_t ws_size,
                              hipStream_t stream) {
    const float* x = (const float*)d_in[0];       // [N, 64]
    const float* w = (const float*)d_in[1];       // [1024, 64]
    float* out = (float*)d_out;                   // [loss | quantized | encodings]

    int*   idxbuf   = (int*)d_ws;                                    // N ints
    float* partials = (float*)((char*)d_ws + (size_t)N_PTS * sizeof(int)); // 1024 f

    float* out_q   = out + 1;
    float* out_enc = out + 1 + (size_t)N_PTS * D_EMB;

    const size_t ldsBytes = (size_t)(W2OFF + K_EMB) * sizeof(float); // ~276 KB

    vq_argmin_kernel<<<256, 512, ldsBytes, stream>>>(x, w, idxbuf);
    vq_quant_loss_kernel<<<1024, 256, 0, stream>>>(x, w, idxbuf, out_q, partials);
    vq_finalize_kernel<<<1, 64, 0, stream>>>(partials, 1024, out);
    vq_encodings_kernel<<<8192, 256, 0, stream>>>(idxbuf, out_enc);
}